// feature_selection_GNN_2473901162531
// MI455X (gfx1250) — compile-verified
//
#include <hip/hip_runtime.h>
#include <hip/hip_bf16.h>
#include <math.h>

// ---------------- problem constants (from reference) ----------------
#define Bv    1024
#define NPGv  116
#define Nn    (Bv * NPGv)       // 118784 nodes
#define DEGv  16
#define Ev    (Nn * DEGv)       // 1900544 edges
#define Finv  116
#define Cc    50
#define K1v   58
#define K2v   29
#define NSNPv 3001
#define NQTv  8
#define NCLSv 4
#define FC1v  500
#define EPSv  1e-5f

// ---------------- workspace layout (float elements) ----------------
constexpr int OFF_H      = 0;
constexpr int OFF_X1     = OFF_H    + Nn * Cc;        // 5,939,200
constexpr int OFF_DEG    = OFF_X1   + Nn * Cc;        // 11,878,400
constexpr int OFF_DINV   = OFF_DEG  + Nn;
constexpr int OFF_REMAP1 = OFF_DINV + Nn;             // int region
constexpr int OFF_X1P    = OFF_REMAP1 + Nn;
constexpr int OFF_H2     = OFF_X1P  + Bv * K1v * Cc;
constexpr int OFF_X2F    = OFF_H2   + Bv * K1v * Cc;
constexpr int OFF_DEG2   = OFF_X2F  + Bv * K1v * Cc;
constexpr int OFF_DINV2  = OFF_DEG2 + Bv * K1v;
constexpr int OFF_REMAP2 = OFF_DINV2 + Bv * K1v;      // int scratch
constexpr int OFF_X2P    = OFF_REMAP2 + Bv * K1v;
constexpr int WS_FLOATS  = OFF_X2P  + Bv * K2v * Cc;  // 22,806,528 floats (~91 MB)

// sub-allocations inside bufB (valid only after pool1 consumed x1):
constexpr int OFF_T1   = OFF_X1;                      // [1024,500]
constexpr int OFF_XX7  = OFF_T1   + Bv * FC1v;        // [1024,700] concat buffer
constexpr int OFF_T3   = OFF_XX7  + Bv * 700;         // [1024,50]
constexpr int OFF_XX50 = OFF_T3   + Bv * Cc;          // [1024,50]
constexpr int OFF_IWM  = OFF_XX50 + Bv * Cc;          // [3001]
constexpr int OFF_BNM1 = OFF_IWM  + NSNPv;
constexpr int OFF_BNV1 = OFF_BNM1 + FC1v;
constexpr int OFF_BNM2 = OFF_BNV1 + FC1v;
constexpr int OFF_BNV2 = OFF_BNM2 + Cc;

typedef __attribute__((ext_vector_type(2))) float v2f;
typedef __attribute__((ext_vector_type(8))) float v8f;

// ---------------- utility kernels ----------------
__global__ void fill_kernel(float* __restrict__ p, int n, float v) {
    int t = blockIdx.x * blockDim.x + threadIdx.x;
    if (t < n) p[t] = v;
}

__global__ void rsqrt_kernel(const float* __restrict__ in, float* __restrict__ out, int n) {
    int t = blockIdx.x * blockDim.x + threadIdx.x;
    if (t < n) out[t] = rsqrtf(in[t]);
}

// ---------------- fp32 WMMA GEMM: D = A[M,K] @ B[K,N] (+bias) ----------------
// 4 waves per block; each wave owns one 16x16 C tile (V_WMMA_F32_16X16X4_F32).
// A 16x4 layout: lane = M row (lane&15); vgpr pair = K 2*hi, 2*hi+1 (hi = lane>>4).
// B 4x16 layout: lane = N col (lane&15); vgpr pair = K rows 2*hi, 2*hi+1.
// C/D layout: vgpr r, lanes0-15 -> (M=r, N=lane), lanes16-31 -> (M=8+r, N=lane-16).
// M % 16 == 0 at every call site. N tail handled by clamping the B column pointer
// (garbage feeds only unstored output columns); K tail = one guarded step.
// Main loop has NO per-lane guards -> unconditional loads, no exec toggling.
__global__ void gemm_wmma_f32(const float* __restrict__ A, int lda,
                              const float* __restrict__ Bm, int ldb,
                              const float* __restrict__ bias,
                              float* __restrict__ D, int ldd,
                              int M, int N, int K, int do_relu) {
    int ntn    = (N + 15) >> 4;
    int ntiles = (M >> 4) * ntn;
    int wave   = threadIdx.x >> 5;                    // 4 waves / block
    int tile   = blockIdx.x * 4 + wave;
    if (tile >= ntiles) return;                        // wave-uniform: EXEC stays all-1s

    int tm = tile / ntn, tn = tile - tm * ntn;
    int m0 = tm << 4, n0 = tn << 4;
    int lane = threadIdx.x & 31;
    int lr = lane & 15, hi = lane >> 4;

    const float* ar = A + (m0 + lr) * lda;            // this lane's A row
    int n = n0 + lr;
    bool ncol = (n < N);
    const float* bc = Bm + (ncol ? n : (N - 1));      // clamped B column

    v8f acc = {};
    if (bias) {
        float bv = bias[ncol ? n : (N - 1)];
        for (int r = 0; r < 8; ++r) acc[r] = bv;
    }

    int K4 = K & ~3;
    #pragma unroll 4
    for (int kb = 0; kb < K4; kb += 4) {
        int ka = kb + 2 * hi;                         // ka+1 <= K4-1 < K: in range
        v2f a, b;
        a.x = ar[ka];
        a.y = ar[ka + 1];
        b.x = bc[ka * ldb];
        b.y = bc[(ka + 1) * ldb];
        acc = __builtin_amdgcn_wmma_f32_16x16x4_f32(
            false, a, false, b, (short)0, acc, false, false);
    }
    if (K4 < K) {                                     // K remainder (1..3), once
        int ka = K4 + 2 * hi;
        v2f a, b;
        a.x = (ka     < K) ? ar[ka]           : 0.0f;
        a.y = (ka + 1 < K) ? ar[ka + 1]       : 0.0f;
        b.x = (ka     < K) ? bc[ka * ldb]     : 0.0f;
        b.y = (ka + 1 < K) ? bc[(ka + 1) * ldb] : 0.0f;
        acc = __builtin_amdgcn_wmma_f32_16x16x4_f32(
            false, a, false, b, (short)0, acc, false, false);
    }

    if (ncol) {
        for (int r = 0; r < 8; ++r) {
            int row = m0 + r + 8 * hi;
            float v = acc[r];
            if (do_relu && v < 0.0f) v = 0.0f;
            D[row * ldd + n] = v;
        }
    }
}

// ---------------- GCN edge kernels ----------------
__global__ void gcn_deg(const int* __restrict__ srcA, const int* __restrict__ dstA, int E,
                        const int* __restrict__ remap, float* __restrict__ deg) {
    int e = blockIdx.x * blockDim.x + threadIdx.x;
    if (e >= E) return;
    int s = srcA[e], d = dstA[e];
    if (remap) { s = remap[s]; d = remap[d]; if (s < 0 || d < 0) return; }
    atomicAdd(&deg[d], 1.0f);
}

// one block of 256 threads handles 4 edges; threadIdx.x&63 indexes the feature
__global__ void gcn_agg(const int* __restrict__ srcA, const int* __restrict__ dstA, int E,
                        const int* __restrict__ remap,
                        const float* __restrict__ h, const float* __restrict__ dinv,
                        float* __restrict__ agg) {
    int e = blockIdx.x * 4 + (threadIdx.x >> 6);
    int f = threadIdx.x & 63;
    if (e >= E || f >= Cc) return;
    int s = srcA[e], d = dstA[e];
    if (remap) { s = remap[s]; d = remap[d]; if (s < 0 || d < 0) return; }
    float coef = dinv[s] * dinv[d];
    atomicAdd(&agg[d * Cc + f], h[s * Cc + f] * coef);
}

__global__ void gcn_fin(const float* __restrict__ agg, const float* __restrict__ h,
                        const float* __restrict__ dinv, const float* __restrict__ bias,
                        float* __restrict__ xout, int nNodes) {
    int t = blockIdx.x * blockDim.x + threadIdx.x;
    if (t >= nNodes * Cc) return;
    int i = t / Cc, f = t - i * Cc;
    float di = dinv[i];
    xout[t] = agg[t] + h[t] * di * di + bias[f];
}

// ---------------- Top-K pooling: one block per graph ----------------
__global__ void topk_pool(const float* __restrict__ xin, const float* __restrict__ p,
                          int npg, int k, int* __restrict__ remap,
                          float* __restrict__ xout) {
    __shared__ float sc[128];
    __shared__ float sinv;
    int g = blockIdx.x;
    int tid = threadIdx.x;
    if (tid == 0) {
        float s = 0.0f;
        for (int i = 0; i < Cc; ++i) s += p[i] * p[i];
        sinv = rsqrtf(s);
    }
    __syncthreads();
    if (tid < npg) {
        const float* row = xin + (g * npg + tid) * Cc;
        float d = 0.0f;
        for (int i = 0; i < Cc; ++i) d += row[i] * p[i];
        sc[tid] = tanhf(d * sinv);
    }
    __syncthreads();
    if (tid < npg) {
        float sn = sc[tid];
        int rank = 0;
        for (int j = 0; j < npg; ++j) {
            float sj = sc[j];
            if (sj > sn || (sj == sn && j < tid)) rank++;
        }
        int node = g * npg + tid;
        if (rank < k) {
            int nd = g * k + rank;
            remap[node] = nd;
            const float* row = xin + node * Cc;
            float* orow = xout + nd * Cc;
            for (int i = 0; i < Cc; ++i) orow[i] = row[i] * sn;
        } else {
            remap[node] = -1;
        }
    }
}

// ---------------- per-graph max/mean readout into xx700[:, 500+off ...] ----------------
__global__ void readout(const float* __restrict__ xp, int k,
                        float* __restrict__ xx700, int off) {
    int t = blockIdx.x * blockDim.x + threadIdx.x;
    if (t >= Bv * Cc) return;
    int g = t / Cc, f = t - g * Cc;
    const float* base = xp + g * k * Cc + f;
    float mx = base[0], sm = base[0];
    for (int j = 1; j < k; ++j) {
        float v = base[j * Cc];
        mx = fmaxf(mx, v);
        sm += v;
    }
    xx700[g * 700 + 500 + off + f]      = mx;
    xx700[g * 700 + 500 + off + Cc + f] = sm / (float)k;
}

// ---------------- SNP input layer ----------------
__global__ void iw_mean(const float* __restrict__ in_w, float* __restrict__ iwm) {
    int j = blockIdx.x * blockDim.x + threadIdx.x;
    if (j >= NSNPv) return;
    float s = 0.0f;
    for (int q = 0; q < NQTv; ++q) s += in_w[q * NSNPv + j];
    iwm[j] = s * (1.0f / NQTv);
}

__global__ void snp_in(const float* __restrict__ x_snp, const float* __restrict__ iwm,
                       float* __restrict__ xs) {
    int j = blockIdx.x * blockDim.x + threadIdx.x;
    int b = blockIdx.y;
    if (j >= NSNPv) return;
    float v = x_snp[b * NSNPv + j] * iwm[j];
    xs[b * NSNPv + j] = v > 0.0f ? v : 0.0f;
}

// ---------------- BatchNorm: block-per-column stats (biased var), then apply ----------------
__global__ void bn_stats(const float* __restrict__ X, int rows, int cols,
                         float* __restrict__ mean, float* __restrict__ vinv) {
    __shared__ float ss[256], sq[256];
    int c = blockIdx.x, tid = threadIdx.x;
    float s = 0.0f, q = 0.0f;
    for (int r = tid; r < rows; r += 256) {
        float v = X[r * cols + c];
        s += v; q += v * v;
    }
    ss[tid] = s; sq[tid] = q;
    __syncthreads();
    for (int w = 128; w > 0; w >>= 1) {
        if (tid < w) { ss[tid] += ss[tid + w]; sq[tid] += sq[tid + w]; }
        __syncthreads();
    }
    if (tid == 0) {
        float m = ss[0] / (float)rows;
        float v = sq[0] / (float)rows - m * m;
        if (v < 0.0f) v = 0.0f;
        mean[c] = m;
        vinv[c] = rsqrtf(v + EPSv);
    }
}

__global__ void bn_apply(const float* __restrict__ X, int rows, int cols,
                         const float* __restrict__ mean, const float* __restrict__ vinv,
                         const float* __restrict__ gam, const float* __restrict__ bet,
                         float* __restrict__ Dst, int dstStride, int do_relu) {
    int t = blockIdx.x * blockDim.x + threadIdx.x;
    if (t >= rows * cols) return;
    int r = t / cols, c = t - r * cols;
    float v = gam[c] * (X[t] - mean[c]) * vinv[c] + bet[c];
    if (do_relu && v < 0.0f) v = 0.0f;
    Dst[r * dstStride + c] = v;
}

// ---------------- small FC heads ----------------
__global__ void fc_reg(const float* __restrict__ xx700, const float* __restrict__ W,
                       const float* __restrict__ bias, float* __restrict__ out) {
    int t = blockIdx.x * blockDim.x + threadIdx.x;
    if (t >= Bv * NQTv) return;
    int b = t / NQTv, o = t - b * NQTv;
    const float* row = xx700 + b * 700;   // first 500 cols = xs2
    float s = bias[o];
    for (int i = 0; i < FC1v; ++i) s += row[i] * W[i * NQTv + o];
    out[t] = s > 0.0f ? s : 0.0f;
}

__global__ void fc_cls(const float* __restrict__ xx50, const float* __restrict__ W,
                       const float* __restrict__ bias, float* __restrict__ out) {
    int b = blockIdx.x * blockDim.x + threadIdx.x;
    if (b >= Bv) return;
    float z[NCLSv];
    for (int o = 0; o < NCLSv; ++o) {
        float s = bias[o];
        for (int i = 0; i < Cc; ++i) s += xx50[b * Cc + i] * W[i * NCLSv + o];
        z[o] = s;
    }
    float m = fmaxf(fmaxf(z[0], z[1]), fmaxf(z[2], z[3]));
    float se = 0.0f;
    for (int o = 0; o < NCLSv; ++o) se += expf(z[o] - m);
    float lse = m + logf(se);
    for (int o = 0; o < NCLSv; ++o) out[Bv * NQTv + b * NCLSv + o] = z[o] - lse;
}

// ---------------- launch ----------------
extern "C" void kernel_launch(void* const* d_in, const int* in_sizes, int n_in,
                              void* d_out, int out_size, void* d_ws, size_t ws_size,
                              hipStream_t stream) {
    const float* x     = (const float*)d_in[0];
    const float* x_snp = (const float*)d_in[1];
    const float* W1    = (const float*)d_in[2];
    const float* b1    = (const float*)d_in[3];
    const float* W2    = (const float*)d_in[4];
    const float* b2    = (const float*)d_in[5];
    const float* p1    = (const float*)d_in[6];
    const float* p2    = (const float*)d_in[7];
    const float* in_w  = (const float*)d_in[8];
    const float* fc1W  = (const float*)d_in[9];
    const float* fc1b  = (const float*)d_in[10];
    const float* bn1g  = (const float*)d_in[11];
    const float* bn1b  = (const float*)d_in[12];
    const float* fc2W  = (const float*)d_in[13];
    const float* fc2b  = (const float*)d_in[14];
    const float* fc3W  = (const float*)d_in[15];
    const float* fc3b  = (const float*)d_in[16];
    const float* bn2g  = (const float*)d_in[17];
    const float* bn2b  = (const float*)d_in[18];
    const float* fc4W  = (const float*)d_in[19];
    const float* fc4b  = (const float*)d_in[20];
    const int*   eidx  = (const int*)d_in[21];
    const int* srcA = eidx;
    const int* dstA = eidx + Ev;

    if (ws_size < (size_t)WS_FLOATS * sizeof(float)) return;

    float* ws   = (float*)d_ws;
    float* h    = ws + OFF_H;
    float* x1   = ws + OFF_X1;     // also agg buffer (in-place finalize)
    float* deg  = ws + OFF_DEG;
    float* dinv = ws + OFF_DINV;
    int*   rm1  = (int*)(ws + OFF_REMAP1);
    float* x1p  = ws + OFF_X1P;
    float* h2   = ws + OFF_H2;
    float* x2f  = ws + OFF_X2F;    // also agg2 buffer
    float* deg2 = ws + OFF_DEG2;
    float* din2 = ws + OFF_DINV2;
    int*   rm2  = (int*)(ws + OFF_REMAP2);
    float* x2p  = ws + OFF_X2P;
    float* xs   = ws + OFF_H;      // reuse (h dead after gcn_fin #1)
    float* t1   = ws + OFF_T1;
    float* xx7  = ws + OFF_XX7;
    float* t3   = ws + OFF_T3;
    float* xx50 = ws + OFF_XX50;
    float* iwm  = ws + OFF_IWM;
    float* bnm1 = ws + OFF_BNM1;
    float* bnv1 = ws + OFF_BNV1;
    float* bnm2 = ws + OFF_BNM2;
    float* bnv2 = ws + OFF_BNV2;
    float* out  = (float*)d_out;

    const int T = 256;
    auto cdiv = [](int a, int b) { return (a + b - 1) / b; };
    auto gemmGrid = [&](int M, int N) { return cdiv((M >> 4) * cdiv(N, 16), 4); };

    // ---- init accumulators ----
    fill_kernel<<<cdiv(Nn, T), T, 0, stream>>>(deg, Nn, 1.0f);                 // +1 self loop
    fill_kernel<<<cdiv(Nn * Cc, T), T, 0, stream>>>(x1, Nn * Cc, 0.0f);        // agg1
    fill_kernel<<<cdiv(Bv * K1v, T), T, 0, stream>>>(deg2, Bv * K1v, 1.0f);    // +1 self loop
    fill_kernel<<<cdiv(Bv * K1v * Cc, T), T, 0, stream>>>(x2f, Bv * K1v * Cc, 0.0f);

    // ---- GCN1: h = x @ W1 ; agg ; x1 = agg + h*dinv^2 + b1 ----
    gemm_wmma_f32<<<gemmGrid(Nn, Cc), 128, 0, stream>>>(
        x, Finv, W1, Cc, nullptr, h, Cc, Nn, Cc, Finv, 0);
    gcn_deg<<<cdiv(Ev, T), T, 0, stream>>>(srcA, dstA, Ev, nullptr, deg);
    rsqrt_kernel<<<cdiv(Nn, T), T, 0, stream>>>(deg, dinv, Nn);
    gcn_agg<<<cdiv(Ev, 4), T, 0, stream>>>(srcA, dstA, Ev, nullptr, h, dinv, x1);
    gcn_fin<<<cdiv(Nn * Cc, T), T, 0, stream>>>(x1, h, dinv, b1, x1, Nn);

    // ---- TopKPool1: 116 -> 58 per graph ----
    topk_pool<<<Bv, 128, 0, stream>>>(x1, p1, NPGv, K1v, rm1, x1p);

    // ---- GCN2: h2 = x1p @ W2 ; masked agg via remap ----
    gemm_wmma_f32<<<gemmGrid(Bv * K1v, Cc), 128, 0, stream>>>(
        x1p, Cc, W2, Cc, nullptr, h2, Cc, Bv * K1v, Cc, Cc, 0);
    gcn_deg<<<cdiv(Ev, T), T, 0, stream>>>(srcA, dstA, Ev, rm1, deg2);
    rsqrt_kernel<<<cdiv(Bv * K1v, T), T, 0, stream>>>(deg2, din2, Bv * K1v);
    gcn_agg<<<cdiv(Ev, 4), T, 0, stream>>>(srcA, dstA, Ev, rm1, h2, din2, x2f);
    gcn_fin<<<cdiv(Bv * K1v * Cc, T), T, 0, stream>>>(x2f, h2, din2, b2, x2f, Bv * K1v);

    // ---- TopKPool2: 58 -> 29 per graph ----
    topk_pool<<<Bv, 128, 0, stream>>>(x2f, p2, K1v, K2v, rm2, x2p);

    // ---- readouts into xx700[:, 500..699] (bufB free now) ----
    readout<<<cdiv(Bv * Cc, T), T, 0, stream>>>(x1p, K1v, xx7, 0);
    readout<<<cdiv(Bv * Cc, T), T, 0, stream>>>(x2p, K2v, xx7, 100);

    // ---- SNP branch ----
    iw_mean<<<cdiv(NSNPv, T), T, 0, stream>>>(in_w, iwm);
    snp_in<<<dim3(cdiv(NSNPv, T), Bv), T, 0, stream>>>(x_snp, iwm, xs);
    gemm_wmma_f32<<<gemmGrid(Bv, FC1v), 128, 0, stream>>>(
        xs, NSNPv, fc1W, FC1v, fc1b, t1, FC1v, Bv, FC1v, NSNPv, 0);
    bn_stats<<<FC1v, 256, 0, stream>>>(t1, Bv, FC1v, bnm1, bnv1);
    bn_apply<<<cdiv(Bv * FC1v, T), T, 0, stream>>>(t1, Bv, FC1v, bnm1, bnv1,
                                                   bn1g, bn1b, xx7, 700, 1);

    // ---- reg head: relu(xs2 @ fc2W + fc2b) -> out[0 .. 8191] ----
    fc_reg<<<cdiv(Bv * NQTv, T), T, 0, stream>>>(xx7, fc2W, fc2b, out);

    // ---- fc3 + BN2 + cls head ----
    gemm_wmma_f32<<<gemmGrid(Bv, Cc), 128, 0, stream>>>(
        xx7, 700, fc3W, Cc, fc3b, t3, Cc, Bv, Cc, 700, 0);
    bn_stats<<<Cc, 256, 0, stream>>>(t3, Bv, Cc, bnm2, bnv2);
    bn_apply<<<cdiv(Bv * Cc, T), T, 0, stream>>>(t3, Bv, Cc, bnm2, bnv2,
                                                 bn2g, bn2b, xx50, Cc, 0);
    fc_cls<<<cdiv(Bv, T), T, 0, stream>>>(xx50, fc4W, fc4b, out);
}